// GNNModel_68204080660970
// MI455X (gfx1250) — compile-verified
//
#include <hip/hip_runtime.h>
#include <hip/hip_bf16.h>
#include <math.h>

// ---------------------------------------------------------------------------
// 2-layer GCN, N=500000 nodes, E=8000000 edges.
// x is [N,1] so layer-1 collapses to a SCALAR edge scatter s[i]; the only
// dense matmul is relu(s*W1+b1) @ W2 ([N,16]x[16,2]), done with
// V_WMMA_F32_16X16X4_F32 on 16-node tiles (W2 zero-padded to 16 columns).
// Node arrays (~2MB) are L2-resident (192MB L2); runtime is bounded by
// streaming the 64MB edge list 3x from HBM @ 23.3 TB/s.
// ---------------------------------------------------------------------------

typedef __attribute__((ext_vector_type(2))) float v2f;
typedef __attribute__((ext_vector_type(4))) float v4f;
typedef __attribute__((ext_vector_type(8))) float v8f;

__global__ __launch_bounds__(256) void k_zero(float* __restrict__ deg,
                                              float* __restrict__ out,
                                              int N) {
  int i = blockIdx.x * blockDim.x + threadIdx.x;
  if (i < N) deg[i] = 0.0f;
  if (i < 2 * N) out[i] = 0.0f;
}

// ---- degree scatter: 4 edges per thread (int4 index loads) ----
__global__ __launch_bounds__(256) void k_degree4(const int* __restrict__ dst,
                                                 float* __restrict__ deg,
                                                 int E4) {
  int i = blockIdx.x * blockDim.x + threadIdx.x;
  if (i < E4) {
    int4 d = ((const int4*)dst)[i];
    unsafeAtomicAdd(&deg[d.x], 1.0f);
    unsafeAtomicAdd(&deg[d.y], 1.0f);
    unsafeAtomicAdd(&deg[d.z], 1.0f);
    unsafeAtomicAdd(&deg[d.w], 1.0f);
  }
}
__global__ __launch_bounds__(256) void k_degree1(const int* __restrict__ dst,
                                                 float* __restrict__ deg,
                                                 int base, int E) {
  int i = base + blockIdx.x * blockDim.x + threadIdx.x;
  if (i < E) unsafeAtomicAdd(&deg[dst[i]], 1.0f);
}

// deg -> dinv (in place) and seed s with the self-loop term x[i]*dinv^2.
__global__ __launch_bounds__(256) void k_dinv_self(float* __restrict__ deg_dinv,
                                                   float* __restrict__ s,
                                                   const float* __restrict__ x,
                                                   int N) {
  int i = blockIdx.x * blockDim.x + threadIdx.x;
  if (i < N) {
    float dv = rsqrtf(deg_dinv[i] + 1.0f);  // +1 = self loop; always > 0
    deg_dinv[i] = dv;
    s[i] = x[i] * dv * dv;
  }
}

// ---- layer-1 edge scatter: s[dst] += x[src]*dinv[src]*dinv[dst] ----
__global__ __launch_bounds__(256) void k_edge1_4(const int* __restrict__ src,
                                                 const int* __restrict__ dst,
                                                 const float* __restrict__ x,
                                                 const float* __restrict__ dinv,
                                                 float* __restrict__ s, int E4) {
  int i = blockIdx.x * blockDim.x + threadIdx.x;
  if (i < E4) {
    int4 u = ((const int4*)src)[i];
    int4 v = ((const int4*)dst)[i];
    unsafeAtomicAdd(&s[v.x], x[u.x] * dinv[u.x] * dinv[v.x]);
    unsafeAtomicAdd(&s[v.y], x[u.y] * dinv[u.y] * dinv[v.y]);
    unsafeAtomicAdd(&s[v.z], x[u.z] * dinv[u.z] * dinv[v.z]);
    unsafeAtomicAdd(&s[v.w], x[u.w] * dinv[u.w] * dinv[v.w]);
  }
}
__global__ __launch_bounds__(256) void k_edge1_1(const int* __restrict__ src,
                                                 const int* __restrict__ dst,
                                                 const float* __restrict__ x,
                                                 const float* __restrict__ dinv,
                                                 float* __restrict__ s,
                                                 int base, int E) {
  int i = base + blockIdx.x * blockDim.x + threadIdx.x;
  if (i < E) {
    int u = src[i], v = dst[i];
    unsafeAtomicAdd(&s[v], x[u] * dinv[u] * dinv[v]);
  }
}

// hw2c[k*N + i] = sum_f relu(s[i]*W1[f]+b1[f]) * W2[f*2+k], k in {0,1}
// One wave per 16-node tile; 4x V_WMMA_F32_16X16X4_F32 cover the K=16
// contraction.  A (f32 16x4): lane L, VGPR j -> K = 2*(L>=16)+j.
// B (4x16, W2 padded with 14 zero cols): VGPR j -> row 2*(L>=16)+j, col L&15.
// D (16x16): lane L, VGPR v -> row (node) v + 8*(L>=16), col (class) L&15.
// All B / W1 / b1 values are loop-invariant: hoisted, loaded branchlessly
// (clamped index + select) so no EXEC toggling inside the tile loop.
// Requires N % 16 == 0 (N = 500000 -> 31250 tiles).
__global__ __launch_bounds__(256) void k_hw2_wmma(const float* __restrict__ s,
                                                  const float* __restrict__ W1,
                                                  const float* __restrict__ b1,
                                                  const float* __restrict__ W2,
                                                  float* __restrict__ hw2c,
                                                  int nTiles, int N) {
  const int lane = threadIdx.x & 31;
  const int m    = lane & 15;   // node-in-tile for A, column for B/D
  const int hi   = lane >> 4;
  const int waveId = (blockIdx.x * blockDim.x + threadIdx.x) >> 5;
  const int nWaves = (gridDim.x * blockDim.x) >> 5;

  // Hoisted per-lane constants (per K-chunk c: elements f0 = 4c+2*hi, f0+1).
  float w1x[4], w1y[4], b1x[4], b1y[4], bmx[4], bmy[4];
  const int mm = (m < 2) ? m : 0;      // clamped -> unconditional loads
  const bool valid = (m < 2);
#pragma unroll
  for (int c = 0; c < 4; ++c) {
    const int f0 = 4 * c + 2 * hi;
    w1x[c] = W1[f0];
    w1y[c] = W1[f0 + 1];
    b1x[c] = b1[f0];
    b1y[c] = b1[f0 + 1];
    float wx = W2[f0 * 2 + mm];
    float wy = W2[(f0 + 1) * 2 + mm];
    bmx[c] = valid ? wx : 0.0f;        // v_cndmask, EXEC stays all-ones
    bmy[c] = valid ? wy : 0.0f;
  }

  for (int t = waveId; t < nTiles; t += nWaves) {
    const int base = t << 4;
    const float sv = s[base + m];
    v8f acc = {};
#pragma unroll
    for (int c = 0; c < 4; ++c) {
      float h0 = fmaf(sv, w1x[c], b1x[c]);
      float h1 = fmaf(sv, w1y[c], b1y[c]);
      v2f a, b;
      a.x = h0 > 0.0f ? h0 : 0.0f;
      a.y = h1 > 0.0f ? h1 : 0.0f;
      b.x = bmx[c];
      b.y = bmy[c];
      acc = __builtin_amdgcn_wmma_f32_16x16x4_f32(
          /*neg_a=*/false, a, /*neg_b=*/false, b,
          /*c_mod=*/(short)0, acc, /*reuse_a=*/false, /*reuse_b=*/false);
    }
    // Lanes {0,1,16,17} hold the two valid D columns: lane (k + 16*hi),
    // VGPR v = class k of node base + v + 8*hi -> contiguous column-major runs.
    if (m < 2) {
      float* p = hw2c + (size_t)m * (size_t)N + (size_t)(base + 8 * hi);
      v4f lo = {acc[0], acc[1], acc[2], acc[3]};
      v4f hv = {acc[4], acc[5], acc[6], acc[7]};
      *(v4f*)(p)     = lo;
      *(v4f*)(p + 4) = hv;
    }
  }
}

// ---- layer-2 edge scatter directly into d_out (row-major [N,2]) ----
__global__ __launch_bounds__(256) void k_edge2_4(const int* __restrict__ src,
                                                 const int* __restrict__ dst,
                                                 const float* __restrict__ dinv,
                                                 const float* __restrict__ hw2c,
                                                 float* __restrict__ out,
                                                 int E4, int N) {
  int i = blockIdx.x * blockDim.x + threadIdx.x;
  if (i < E4) {
    int4 u = ((const int4*)src)[i];
    int4 v = ((const int4*)dst)[i];
    float n0 = dinv[u.x] * dinv[v.x];
    float n1 = dinv[u.y] * dinv[v.y];
    float n2 = dinv[u.z] * dinv[v.z];
    float n3 = dinv[u.w] * dinv[v.w];
    unsafeAtomicAdd(&out[2 * v.x],     hw2c[u.x] * n0);
    unsafeAtomicAdd(&out[2 * v.x + 1], hw2c[N + u.x] * n0);
    unsafeAtomicAdd(&out[2 * v.y],     hw2c[u.y] * n1);
    unsafeAtomicAdd(&out[2 * v.y + 1], hw2c[N + u.y] * n1);
    unsafeAtomicAdd(&out[2 * v.z],     hw2c[u.z] * n2);
    unsafeAtomicAdd(&out[2 * v.z + 1], hw2c[N + u.z] * n2);
    unsafeAtomicAdd(&out[2 * v.w],     hw2c[u.w] * n3);
    unsafeAtomicAdd(&out[2 * v.w + 1], hw2c[N + u.w] * n3);
  }
}
__global__ __launch_bounds__(256) void k_edge2_1(const int* __restrict__ src,
                                                 const int* __restrict__ dst,
                                                 const float* __restrict__ dinv,
                                                 const float* __restrict__ hw2c,
                                                 float* __restrict__ out,
                                                 int base, int E, int N) {
  int i = base + blockIdx.x * blockDim.x + threadIdx.x;
  if (i < E) {
    int u = src[i], v = dst[i];
    float nrm = dinv[u] * dinv[v];
    unsafeAtomicAdd(&out[2 * v],     hw2c[u] * nrm);
    unsafeAtomicAdd(&out[2 * v + 1], hw2c[N + u] * nrm);
  }
}

// Self-loop term + bias + numerically stable 2-class log-softmax, in place.
__global__ __launch_bounds__(256) void k_finalize(float* __restrict__ out,
                                                  const float* __restrict__ hw2c,
                                                  const float* __restrict__ dinv,
                                                  const float* __restrict__ b2,
                                                  int N) {
  int i = blockIdx.x * blockDim.x + threadIdx.x;
  if (i < N) {
    float dv = dinv[i];
    float sl = dv * dv;
    float z0 = out[2 * i]     + hw2c[i]     * sl + b2[0];
    float z1 = out[2 * i + 1] + hw2c[N + i] * sl + b2[1];
    float mx = fmaxf(z0, z1), mn = fminf(z0, z1);
    float lse = mx + log1pf(expf(mn - mx));
    float2 r;
    r.x = z0 - lse;
    r.y = z1 - lse;
    *(float2*)(out + 2 * i) = r;
  }
}

extern "C" void kernel_launch(void* const* d_in, const int* in_sizes, int n_in,
                              void* d_out, int out_size, void* d_ws, size_t ws_size,
                              hipStream_t stream) {
  const float* x  = (const float*)d_in[0];
  const int*   ei = (const int*)d_in[1];
  const float* W1 = (const float*)d_in[2];
  const float* b1 = (const float*)d_in[3];
  const float* W2 = (const float*)d_in[4];
  const float* b2 = (const float*)d_in[5];

  const int N = in_sizes[0];      // 500000 (divisible by 16)
  const int E = in_sizes[1] / 2;  // 8000000 (divisible by 4)
  const int* src = ei;
  const int* dst = ei + E;

  // Workspace layout: [deg->dinv : N][s : N][hw2c : 2N]  (8 MB total)
  float* ws   = (float*)d_ws;
  float* dinv = ws;                    // deg then, in place, rsqrt(deg+1)
  float* s    = ws + (size_t)N;
  float* hw2c = ws + 2 * (size_t)N;    // column-major [2][N]
  float* out  = (float*)d_out;         // doubles as layer-2 accumulator

  const int T = 256;
  const int E4    = E >> 2;            // vectorized edge quads
  const int Etail = E & 3;
  const int tBase = E4 << 2;
  const int gN  = (N + T - 1) / T;
  const int g2N = (2 * N + T - 1) / T;
  const int gE4 = (E4 + T - 1) / T;
  const int nTiles = N / 16;           // 31250
  const int gW = (nTiles + (T / 32) - 1) / (T / 32);

  k_zero<<<g2N, T, 0, stream>>>(dinv, out, N);

  k_degree4<<<gE4, T, 0, stream>>>(dst, dinv, E4);
  if (Etail) k_degree1<<<1, T, 0, stream>>>(dst, dinv, tBase, E);

  k_dinv_self<<<gN, T, 0, stream>>>(dinv, s, x, N);

  k_edge1_4<<<gE4, T, 0, stream>>>(src, dst, x, dinv, s, E4);
  if (Etail) k_edge1_1<<<1, T, 0, stream>>>(src, dst, x, dinv, s, tBase, E);

  k_hw2_wmma<<<gW, T, 0, stream>>>(s, W1, b1, W2, hw2c, nTiles, N);

  k_edge2_4<<<gE4, T, 0, stream>>>(src, dst, dinv, hw2c, out, E4, N);
  if (Etail) k_edge2_1<<<1, T, 0, stream>>>(src, dst, dinv, hw2c, out, tBase, E, N);

  k_finalize<<<gN, T, 0, stream>>>(out, hw2c, dinv, b2, N);
}